// MakeCutouts_75823352644348
// MI455X (gfx1250) — compile-verified
//
#include <hip/hip_runtime.h>

// MakeCutouts: out[n,b,c,i,j] = x[b, c, oy[n] + i*s[n]/224, ox[n] + j*s[n]/224]
// x = (16,3,512,512) f32 (50MB, L2-resident), out = (512,3,224,224) f32 (308MB).
// Pure gather, 0 FLOPs -> no matrix op to map to WMMA; optimize data movement:
//   * grid.y = (n,b,c) -> table lookups + base addresses are uniform (SALU/s_load)
//   * per-block LDS table of the shared quotient q[i] = i*s/224 (one int div per
//     setup lane); main phase reads 4 x-indices with one ds_load_b128
//   * 4 dword gather loads (stride s/224 in {1,2}, L2 hits), one NT b128 store
//   * grid.x * 256 == 224*56 exactly -> no bounds checks

typedef float v4f __attribute__((ext_vector_type(4)));
typedef int   v4i __attribute__((ext_vector_type(4)));

#define CUT_SZ   224
#define CUT_N    32
#define B_DIM    16
#define C_DIM    3
#define H_DIM    512
#define W_DIM    512
#define JCHUNKS  (CUT_SZ / 4)              // 56 float4 chunks per row
#define PLANE    (CUT_SZ * JCHUNKS)        // 12544 chunks per (n,b,c) plane

__global__ __launch_bounds__(256) void make_cutouts_kernel(
    const float* __restrict__ x,
    const int*   __restrict__ sizes,
    const int*   __restrict__ offsetx,
    const int*   __restrict__ offsety,
    float*       __restrict__ out)
{
    __shared__ __align__(16) int xs[CUT_SZ];  // ox + j*s/224
    __shared__ int ys[CUT_SZ];                // oy + i*s/224

    // (n,b,c) from blockIdx.y — all uniform, SALU only.
    const int nbc = blockIdx.y;        // ((n*16 + b)*3 + c)
    const int c   = nbc % C_DIM;
    const int t   = nbc / C_DIM;
    const int b   = t % B_DIM;
    const int n   = t / B_DIM;

    const int s  = sizes[n];           // uniform address -> s_load
    const int ox = offsetx[n];
    const int oy = offsety[n];

    const int tidx = threadIdx.x;
    if (tidx < CUT_SZ) {
        int q = (tidx * s) / CUT_SZ;   // one magic-mul division per setup lane
        xs[tidx] = ox + q;
        ys[tidx] = oy + q;
    }
    __syncthreads();

    const int chunk = blockIdx.x * 256 + tidx;   // 0..12543, exact cover
    const int j4 = chunk % JCHUNKS;
    const int i  = chunk / JCHUNKS;

    const int y = ys[i];                               // ds_load_b32 (broadcast)
    const float* __restrict__ src =
        x + (((b * C_DIM + c) * H_DIM + y) * W_DIM);   // uniform base + y*512

    const v4i xi = *(const v4i*)(xs + j4 * 4);         // ds_load_b128, 16B aligned

    __builtin_prefetch(src + xi.x, 0, 3);              // gfx1250 global_prefetch_b8

    v4f v;
    v.x = src[xi.x];
    v.y = src[xi.y];
    v.z = src[xi.z];
    v.w = src[xi.w];

    // Output plane for this (n,b,c) is contiguous: nbc*12544 float4 chunks in.
    __builtin_nontemporal_store(v, (v4f*)out + (nbc * PLANE + chunk));
}

extern "C" void kernel_launch(void* const* d_in, const int* in_sizes, int n_in,
                              void* d_out, int out_size, void* d_ws, size_t ws_size,
                              hipStream_t stream) {
    const float* x       = (const float*)d_in[0];
    const int*   sizes   = (const int*)d_in[1];
    const int*   offsetx = (const int*)d_in[2];
    const int*   offsety = (const int*)d_in[3];
    float*       out     = (float*)d_out;

    dim3 grid(PLANE / 256, CUT_N * B_DIM * C_DIM, 1);  // (49, 1536)
    make_cutouts_kernel<<<grid, dim3(256, 1, 1), 0, stream>>>(
        x, sizes, offsetx, offsety, out);
}